// SpotifyModel_23716809409278
// MI455X (gfx1250) — compile-verified
//
#include <hip/hip_runtime.h>
#include <hip/hip_bf16.h>
#include <math.h>

typedef __attribute__((ext_vector_type(2))) float v2f;
typedef __attribute__((ext_vector_type(8))) float v8f;

#define FEAT      384          // 3 * 128
#define N_CTX     4096
#define N_POS     2048
#define N_NEG     8192
#define N_ALL     (N_CTX + N_POS + N_NEG)
#define MAX_TRK   100000
#define MAX_ALB   100000
#define LDA       388          // padded LDS row stride (388 % 64 == 4 -> conflict-free)

// ---------------------------------------------------------------------------
// Kernel 1: gather embeddings into workspace (row-major [row][384]) + L2 norms
// One 128-thread block per output row; thread t covers feature t of each table.
// ---------------------------------------------------------------------------
__global__ __launch_bounds__(128) void gather_l2_kernel(
    const int* __restrict__ tc, const int* __restrict__ ac, const int* __restrict__ rc,
    const int* __restrict__ tn, const int* __restrict__ an, const int* __restrict__ rn,
    const int* __restrict__ tg, const int* __restrict__ ag, const int* __restrict__ rg,
    const float* __restrict__ track_tab,
    const float* __restrict__ album_tab,
    const float* __restrict__ artist_tab,
    float* __restrict__ ws,           // [N_ALL][384]
    float* __restrict__ l2_out)       // [N_ALL]
{
    const int row = blockIdx.x;
    const int t   = threadIdx.x;

    int track, album, artist;
    if (row < N_CTX) {
        track = tc[row]; album = ac[row]; artist = rc[row];
    } else if (row < N_CTX + N_POS) {
        const int i = row - N_CTX;
        track = tn[i]; album = an[i]; artist = rn[i];
    } else {
        const int i = row - (N_CTX + N_POS);
        track = tg[i]; album = ag[i]; artist = rg[i];
    }

    const float e0 = track_tab [(size_t)(track % MAX_TRK) * 128 + t];
    const float e1 = album_tab [(size_t)(album % MAX_ALB) * 128 + t];
    const float e2 = artist_tab[(size_t)artist            * 128 + t];

    float* wrow = ws + (size_t)row * FEAT;
    wrow[t]       = e0;
    wrow[128 + t] = e1;
    wrow[256 + t] = e2;

    float ss = e0 * e0 + e1 * e1 + e2 * e2;
    #pragma unroll
    for (int off = 16; off > 0; off >>= 1)
        ss += __shfl_down(ss, off, 32);

    __shared__ float part[4];
    if ((t & 31) == 0) part[t >> 5] = ss;
    __syncthreads();
    if (t == 0)
        l2_out[row] = sqrtf(part[0] + part[1] + part[2] + part[3]);
}

// ---------------------------------------------------------------------------
// Kernel 2: rowwise  out[m] = max_n ( A[m,:] . ctx[n,:] )  via f32 WMMA.
// Block = 256 threads (8 waves), 16 rows of A per block staged in LDS.
// Each wave sweeps n-tiles (16 cols) strided by 8; 96 chained
// V_WMMA_F32_16X16X4_F32 per tile (K = 384).
// ---------------------------------------------------------------------------
__global__ __launch_bounds__(256) void gemm_max_kernel(
    const float* __restrict__ A,      // [rows][384]
    const float* __restrict__ ctx,    // [N_CTX][384]
    float* __restrict__ out)          // [rows]
{
    __shared__ float aTile[16 * LDA];
    __shared__ float red[8][16];

    const int tid  = threadIdx.x;
    const int wave = tid >> 5;
    const int lane = tid & 31;
    const int r    = lane & 15;   // row (A frag) / col (B frag) within tile
    const int kh   = lane >> 4;   // which K-halfpair this lane holds

    const int rowbase = blockIdx.x * 16;

    // Stage the 16x384 A block into LDS (padded stride).
    const float* aBlk = A + (size_t)rowbase * FEAT;
    for (int i = tid; i < 16 * FEAT; i += 256) {
        const int rr = i / FEAT;
        const int cc = i - rr * FEAT;
        aTile[rr * LDA + cc] = aBlk[i];
    }
    __syncthreads();

    v8f maxfrag;
    #pragma unroll
    for (int i = 0; i < 8; ++i) maxfrag[i] = -__builtin_inff();

    const float* aLane = &aTile[r * LDA + 2 * kh];

    for (int nt = wave; nt < (N_CTX / 16); nt += 8) {
        const float* bLane = ctx + ((size_t)(nt * 16 + r)) * FEAT + 2 * kh;
        v8f acc = {};
        #pragma unroll 8
        for (int k = 0; k < FEAT; k += 4) {
            const v2f a2 = *(const v2f*)(aLane + k);   // A[r][k+2h], A[r][k+2h+1]
            const v2f b2 = *(const v2f*)(bLane + k);   // ctx[col+r][k+2h..+1]
            acc = __builtin_amdgcn_wmma_f32_16x16x4_f32(
                false, a2, false, b2, (short)0, acc, false, false);
        }
        #pragma unroll
        for (int i = 0; i < 8; ++i) maxfrag[i] = fmaxf(maxfrag[i], acc[i]);
    }

    // Max over N: D layout -> N = lane%16, so xor-reduce inside each 16-lane half.
    #pragma unroll
    for (int off = 1; off <= 8; off <<= 1) {
        #pragma unroll
        for (int i = 0; i < 8; ++i) {
            const float o = __shfl_xor(maxfrag[i], off, 32);
            maxfrag[i] = fmaxf(maxfrag[i], o);
        }
    }
    // lane 0 holds rows 0..7 (vgpr i -> row i); lane 16 holds rows 8..15.
    if (r == 0) {
        const int rb = kh * 8;
        #pragma unroll
        for (int i = 0; i < 8; ++i) red[wave][rb + i] = maxfrag[i];
    }
    __syncthreads();

    if (tid < 16) {
        float m = red[0][tid];
        #pragma unroll
        for (int w = 1; w < 8; ++w) m = fmaxf(m, red[w][tid]);
        out[rowbase + tid] = m;
    }
}

// ---------------------------------------------------------------------------
extern "C" void kernel_launch(void* const* d_in, const int* in_sizes, int n_in,
                              void* d_out, int out_size, void* d_ws, size_t ws_size,
                              hipStream_t stream) {
    const int* tc = (const int*)d_in[0];
    const int* ac = (const int*)d_in[1];
    const int* rc = (const int*)d_in[2];
    const int* tn = (const int*)d_in[3];
    const int* an = (const int*)d_in[4];
    const int* rn = (const int*)d_in[5];
    const int* tg = (const int*)d_in[6];
    const int* ag = (const int*)d_in[7];
    const int* rg = (const int*)d_in[8];
    const float* track_tab  = (const float*)d_in[9];
    const float* album_tab  = (const float*)d_in[10];
    const float* artist_tab = (const float*)d_in[11];

    float* ws  = (float*)d_ws;
    float* ctx = ws;                                   // [4096][384]
    float* nxt = ws + (size_t)N_CTX * FEAT;            // [2048][384]
    float* neg = ws + (size_t)(N_CTX + N_POS) * FEAT;  // [8192][384]

    float* out     = (float*)d_out;
    float* pos_aff = out;                  // [2048]
    float* neg_aff = out + N_POS;          // [8192]
    float* l2_out  = out + N_POS + N_NEG;  // [14336]

    gather_l2_kernel<<<N_ALL, 128, 0, stream>>>(
        tc, ac, rc, tn, an, rn, tg, ag, rg,
        track_tab, album_tab, artist_tab, ws, l2_out);

    gemm_max_kernel<<<N_POS / 16, 256, 0, stream>>>(nxt, ctx, pos_aff);
    gemm_max_kernel<<<N_NEG / 16, 256, 0, stream>>>(neg, ctx, neg_aff);
}